// ClassicalKernalAnsatz_65481071403856
// MI455X (gfx1250) — compile-verified
//
#include <hip/hip_runtime.h>
#include <math.h>

typedef __attribute__((ext_vector_type(2))) float v2f;
typedef __attribute__((ext_vector_type(8))) float v8f;

#define NROWS 8192
#define DDIM  64
#define GAMMA 1.0f

// ---------------------------------------------------------------------------
// Kernel 1: squared row norms of x and y into ws[0..2*NROWS)
// ---------------------------------------------------------------------------
__global__ __launch_bounds__(256) void sqnorm_kernel(const float* __restrict__ x,
                                                     const float* __restrict__ y,
                                                     float* __restrict__ sq) {
    int i = blockIdx.x * blockDim.x + threadIdx.x;   // 0 .. 2*NROWS-1
    const float* src = (i < NROWS) ? x : y;
    int row = (i < NROWS) ? i : (i - NROWS);
    const float4* p = (const float4*)(src + (size_t)row * DDIM);
    float s = 0.0f;
#pragma unroll
    for (int j = 0; j < DDIM / 4; ++j) {
        float4 v = p[j];
        s = fmaf(v.x, v.x, s);
        s = fmaf(v.y, v.y, s);
        s = fmaf(v.z, v.z, s);
        s = fmaf(v.w, v.w, s);
    }
    sq[i] = s;
}

// ---------------------------------------------------------------------------
// Kernel 2: RBF Gram matrix via V_WMMA_F32_16X16X4_F32.
// Block = 256 threads = 8 waves; each wave computes a 32(M) x 64(N) strip
// (2 M-tiles x 4 N-tiles = 8 f32 accumulators, 128 WMMAs per wave).
// Grid = (8192/64, 8192/256).
// ---------------------------------------------------------------------------
__global__ __launch_bounds__(256) void rbf_wmma_kernel(const float* __restrict__ x,
                                                       const float* __restrict__ y,
                                                       const float* __restrict__ sq,
                                                       float* __restrict__ out) {
    const int lane = threadIdx.x & 31;
    const int wave = threadIdx.x >> 5;
    const int half = lane >> 4;      // 0: lanes 0-15, 1: lanes 16-31
    const int l16  = lane & 15;

    const int base_m = blockIdx.y * 256 + wave * 32;
    const int base_n = blockIdx.x * 64;

    // A fragment sources: x[base_m + 16*mt + l16][k0 + 2*half + {0,1}]
    const float* xp = x + (size_t)(base_m + l16) * DDIM + 2 * half;
    // B fragment sources: y[base_n + 16*nt + l16][k0 + 2*half + {0,1}]
    const float* yp = y + (size_t)(base_n + l16) * DDIM + 2 * half;

    v8f c00 = {}, c01 = {}, c02 = {}, c03 = {};
    v8f c10 = {}, c11 = {}, c12 = {}, c13 = {};

#pragma unroll
    for (int s = 0; s < DDIM / 4; ++s) {           // 16 k-steps of K=4
        v2f a0 = *(const v2f*)(xp + 4 * s);
        v2f a1 = *(const v2f*)(xp + 16 * DDIM + 4 * s);
        v2f b0 = *(const v2f*)(yp + 4 * s);
        v2f b1 = *(const v2f*)(yp + 16 * DDIM + 4 * s);
        v2f b2 = *(const v2f*)(yp + 32 * DDIM + 4 * s);
        v2f b3 = *(const v2f*)(yp + 48 * DDIM + 4 * s);
        c00 = __builtin_amdgcn_wmma_f32_16x16x4_f32(false, a0, false, b0, (short)0, c00, false, false);
        c01 = __builtin_amdgcn_wmma_f32_16x16x4_f32(false, a0, false, b1, (short)0, c01, false, false);
        c02 = __builtin_amdgcn_wmma_f32_16x16x4_f32(false, a0, false, b2, (short)0, c02, false, false);
        c03 = __builtin_amdgcn_wmma_f32_16x16x4_f32(false, a0, false, b3, (short)0, c03, false, false);
        c10 = __builtin_amdgcn_wmma_f32_16x16x4_f32(false, a1, false, b0, (short)0, c10, false, false);
        c11 = __builtin_amdgcn_wmma_f32_16x16x4_f32(false, a1, false, b1, (short)0, c11, false, false);
        c12 = __builtin_amdgcn_wmma_f32_16x16x4_f32(false, a1, false, b2, (short)0, c12, false, false);
        c13 = __builtin_amdgcn_wmma_f32_16x16x4_f32(false, a1, false, b3, (short)0, c13, false, false);
    }

    // Epilogue: sq_norm = xs + ys - 2*cross; out = exp(-gamma * max(sq_norm, 0))
    const float* ysq = sq + NROWS;
    float ys0 = ysq[base_n +  0 + l16];
    float ys1 = ysq[base_n + 16 + l16];
    float ys2 = ysq[base_n + 32 + l16];
    float ys3 = ysq[base_n + 48 + l16];

#pragma unroll
    for (int v = 0; v < 8; ++v) {
        int   row0 = base_m + v + 8 * half;        // M-tile 0
        int   row1 = row0 + 16;                    // M-tile 1
        float xs0  = sq[row0];
        float xs1  = sq[row1];
        float* orow0 = out + (size_t)row0 * NROWS + base_n + l16;
        float* orow1 = out + (size_t)row1 * NROWS + base_n + l16;

        float d00 = fmaxf(xs0 + ys0 - 2.0f * c00[v], 0.0f);
        float d01 = fmaxf(xs0 + ys1 - 2.0f * c01[v], 0.0f);
        float d02 = fmaxf(xs0 + ys2 - 2.0f * c02[v], 0.0f);
        float d03 = fmaxf(xs0 + ys3 - 2.0f * c03[v], 0.0f);
        float d10 = fmaxf(xs1 + ys0 - 2.0f * c10[v], 0.0f);
        float d11 = fmaxf(xs1 + ys1 - 2.0f * c11[v], 0.0f);
        float d12 = fmaxf(xs1 + ys2 - 2.0f * c12[v], 0.0f);
        float d13 = fmaxf(xs1 + ys3 - 2.0f * c13[v], 0.0f);

        __builtin_nontemporal_store(__expf(-GAMMA * d00), orow0 +  0);
        __builtin_nontemporal_store(__expf(-GAMMA * d01), orow0 + 16);
        __builtin_nontemporal_store(__expf(-GAMMA * d02), orow0 + 32);
        __builtin_nontemporal_store(__expf(-GAMMA * d03), orow0 + 48);
        __builtin_nontemporal_store(__expf(-GAMMA * d10), orow1 +  0);
        __builtin_nontemporal_store(__expf(-GAMMA * d11), orow1 + 16);
        __builtin_nontemporal_store(__expf(-GAMMA * d12), orow1 + 32);
        __builtin_nontemporal_store(__expf(-GAMMA * d13), orow1 + 48);
    }
}

// ---------------------------------------------------------------------------
extern "C" void kernel_launch(void* const* d_in, const int* in_sizes, int n_in,
                              void* d_out, int out_size, void* d_ws, size_t ws_size,
                              hipStream_t stream) {
    const float* x = (const float*)d_in[0];
    const float* y = (const float*)d_in[1];
    float* out = (float*)d_out;
    float* sq  = (float*)d_ws;   // 2*8192 floats = 64 KB

    sqnorm_kernel<<<(2 * NROWS) / 256, 256, 0, stream>>>(x, y, sq);

    dim3 grid(NROWS / 64, NROWS / 256);
    rbf_wmma_kernel<<<grid, 256, 0, stream>>>(x, y, sq, out);
}